// MultiFeatureGrouping_56676388438355
// MI455X (gfx1250) — compile-verified
//
#include <hip/hip_runtime.h>
#include <math.h>

typedef float v8f __attribute__((ext_vector_type(8)));
typedef float v2f __attribute__((ext_vector_type(2)));

#define NB 4            // batch
#define Q  9216         // 96*96 tokens
#define CH 256          // channels (S)
#define BT (NB*Q)       // 36864 rows
#define IMG_B ((size_t)CH*Q) // 2359296 floats per batch of NCHW tensor

// ---------------------------------------------------------------------------
// positional MLP: out[t][c] = b2[c] + sum_h relu(b1[h] + grid(t).w1[h])*w2[c][h]
// grid(t) = (i/95, j/95, 1-i/95, 1-j/95), t = i*96 + j
// ---------------------------------------------------------------------------
__global__ void pos_mlp_kernel(const float* __restrict__ w1, const float* __restrict__ b1,
                               const float* __restrict__ w2, const float* __restrict__ b2,
                               float* __restrict__ out) {
    int t = blockIdx.x;
    int i = t / 96, j = t % 96;
    float g0 = i * (1.0f / 95.0f);
    float g1 = j * (1.0f / 95.0f);
    __shared__ float hid[64];
    int tid = threadIdx.x;
    if (tid < 64) {
        float a = b1[tid] + w1[tid*4+0]*g0 + w1[tid*4+1]*g1
                + w1[tid*4+2]*(1.0f-g0) + w1[tid*4+3]*(1.0f-g1);
        hid[tid] = fmaxf(a, 0.0f);
    }
    __syncthreads();
    float acc = b2[tid];
    const float* wr = w2 + (size_t)tid * 64;
#pragma unroll 8
    for (int h = 0; h < 64; ++h) acc += hid[h] * wr[h];
    out[(size_t)t * CH + tid] = acc;
}

// Wsum[o] = sum_c Wconv[o][c]
__global__ void wsum_kernel(const float* __restrict__ Wconv, float* __restrict__ Wsum) {
    int o = threadIdx.x;
    float s = 0.0f;
    const float* r = Wconv + (size_t)o * CH;
    for (int c = 0; c < CH; ++c) s += r[c];
    Wsum[o] = s;
}

// P[o][t] = sum_c Wconv[o][c] * dec_pos[c*36 + t/256][t%256]
__global__ void pcomp_kernel(const float* __restrict__ Wconv, const float* __restrict__ dec_pos,
                             float* __restrict__ P) {
    int o   = blockIdx.y;
    int bx  = blockIdx.x;          // t/256
    int tid = threadIdx.x;         // t%256
    int t = bx * 256 + tid;
    const float* wr = Wconv + (size_t)o * CH;
    float acc = 0.0f;
    for (int c = 0; c < CH; ++c)
        acc += wr[c] * dec_pos[((size_t)(c*36 + bx)) * CH + tid];
    P[(size_t)o * Q + t] = acc;
}

// ---------------------------------------------------------------------------
// Fused (x [+ gather(slots1_nhwc)] [+ pos]) -> LayerNorm -> out   (256 ch/row)
// ---------------------------------------------------------------------------
template<bool GATHER, bool POS>
__global__ void ln_kernel(const float* __restrict__ in1, const float* __restrict__ in2,
                          const float* __restrict__ pos,
                          const float* __restrict__ g, const float* __restrict__ bta,
                          float* __restrict__ out) {
    int row = blockIdx.x;
    int c = threadIdx.x;
    size_t base = (size_t)row * CH + c;
    float x = in1[base];
    if (GATHER) {
        int bb = row / Q, tok = row % Q;
        int ql = tok * CH + c;            // NCHW flat offset within batch
        int ch = ql / Q, t = ql % Q;
        x += in2[((size_t)bb * Q + t) * CH + ch];
    }
    if (POS) x += pos[(size_t)(row % Q) * CH + c];
    __shared__ float red[256];
    red[c] = x; __syncthreads();
    for (int s = 128; s; s >>= 1) { if (c < s) red[c] += red[c + s]; __syncthreads(); }
    float mean = red[0] * (1.0f / 256.0f); __syncthreads();
    float d = x - mean;
    red[c] = d * d; __syncthreads();
    for (int s = 128; s; s >>= 1) { if (c < s) red[c] += red[c + s]; __syncthreads(); }
    float var = red[0] * (1.0f / 256.0f);
    out[base] = d * rsqrtf(var + 1e-5f) * g[c] + bta[c];
}

// ---------------------------------------------------------------------------
// WMMA GEMM: C[M,256] = act(A[M,256] @ W[256,256]^T + bias)
// 8 waves/block cover 64 rows x 256 cols; wave tile = 32 rows x 64 cols:
// per k-step 2 A frags + 4 B frags feed 8 v_wmma_f32_16x16x4_f32 (0.75 ld/wmma)
// ---------------------------------------------------------------------------
template<int RELU, int BIAS>
__global__ void gemm_kernel(const float* __restrict__ A, const float* __restrict__ W,
                            const float* __restrict__ bias, float* __restrict__ C) {
    const int K = 256;
    int lane = threadIdx.x & 31, wave = threadIdx.x >> 5;
    int rstrip = wave & 1, ngrp = wave >> 1;
    int m0 = blockIdx.x * 64 + rstrip * 32;
    int n0 = ngrp * 64;
    int lm = lane & 15;
    int lk = (lane >> 4) * 2;
    v8f acc[2][4] = {};
    const float* Ar0 = A + (size_t)(m0 + lm) * K;
    const float* Ar1 = A + (size_t)(m0 + 16 + lm) * K;
    const float* W0  = W + (size_t)(n0 + lm) * K;
    for (int kb = 0; kb < K; kb += 4) {
        v2f a0 = *(const v2f*)(Ar0 + kb + lk);
        v2f a1 = *(const v2f*)(Ar1 + kb + lk);
#pragma unroll
        for (int nt = 0; nt < 4; ++nt) {
            v2f b = *(const v2f*)(W0 + (size_t)nt * 16 * K + kb + lk);
            acc[0][nt] = __builtin_amdgcn_wmma_f32_16x16x4_f32(
                false, a0, false, b, (short)0, acc[0][nt], false, false);
            acc[1][nt] = __builtin_amdgcn_wmma_f32_16x16x4_f32(
                false, a1, false, b, (short)0, acc[1][nt], false, false);
        }
    }
    int mq = (lane >> 4) * 8;
#pragma unroll
    for (int ms = 0; ms < 2; ++ms) {
        int mbase = m0 + ms * 16 + mq;
#pragma unroll
        for (int nt = 0; nt < 4; ++nt) {
            int n = n0 + nt * 16 + lm;
            float bv = BIAS ? bias[n] : 0.0f;
#pragma unroll
            for (int v = 0; v < 8; ++v) {
                float x = acc[ms][nt][v] + bv;
                if (RELU) x = fmaxf(x, 0.0f);
                C[(size_t)(mbase + v) * 256 + n] = x;
            }
        }
    }
}

// slots[n][c] = mu[n%ns][c] + exp(log_sigma[n%ns][c]) * noise[n][c]
__global__ void slots_init_kernel(const float* __restrict__ mu, const float* __restrict__ ls,
                                  const float* __restrict__ noise, float* __restrict__ slots,
                                  int ns) {
    int n = blockIdx.x, c = threadIdx.x;
    int j = n % ns;
    slots[(size_t)n * CH + c] = mu[(size_t)j * CH + c]
        + __expf(ls[(size_t)j * CH + c]) * noise[(size_t)n * CH + c];
}

// q = LN(slots; ns_g, ns_b) @ Wq^T * 256^-0.5
__global__ void slot_q_kernel(const float* __restrict__ slots,
                              const float* __restrict__ g, const float* __restrict__ bta,
                              const float* __restrict__ Wq, float* __restrict__ q) {
    int n = blockIdx.x, c = threadIdx.x;
    float x = slots[(size_t)n * CH + c];
    __shared__ float red[256];
    __shared__ float lnv[256];
    red[c] = x; __syncthreads();
    for (int s = 128; s; s >>= 1) { if (c < s) red[c] += red[c + s]; __syncthreads(); }
    float mean = red[0] * (1.0f / 256.0f); __syncthreads();
    float d = x - mean;
    red[c] = d * d; __syncthreads();
    for (int s = 128; s; s >>= 1) { if (c < s) red[c] += red[c + s]; __syncthreads(); }
    float var = red[0] * (1.0f / 256.0f);
    lnv[c] = d * rsqrtf(var + 1e-5f) * g[c] + bta[c];
    __syncthreads();
    float acc = 0.0f;
    const float* wr = Wq + (size_t)c * CH;
#pragma unroll 8
    for (int i = 0; i < CH; ++i) acc += lnv[i] * wr[i];
    q[(size_t)n * CH + c] = acc * 0.0625f;  // 1/sqrt(256)
}

// attn[b,tok,j] = softmax_j(k[b,tok,:] . q[b,j,:]) + 1e-8; colsum += attn
__global__ void attn_logits_kernel(const float* __restrict__ kbuf, const float* __restrict__ q,
                                   float* __restrict__ attn, float* __restrict__ colsum,
                                   int ns) {
    int wave = threadIdx.x >> 5, lane = threadIdx.x & 31;
    int bb = blockIdx.y;
    int tok = blockIdx.x * 8 + wave;
    const float* krow = kbuf + ((size_t)bb * Q + tok) * CH;
    __shared__ float aL[8][8];
    float kv[8];
#pragma unroll
    for (int u = 0; u < 8; ++u) kv[u] = krow[lane + 32 * u];
    for (int j = 0; j < ns; ++j) {
        const float* qrow = q + ((size_t)bb * ns + j) * CH;
        float p = 0.0f;
#pragma unroll
        for (int u = 0; u < 8; ++u) p += kv[u] * qrow[lane + 32 * u];
        for (int off = 16; off; off >>= 1) p += __shfl_xor(p, off, 32);
        if (lane == 0) aL[wave][j] = p;
    }
    if (lane == 0) {
        float mx = -3.0e38f;
        for (int j = 0; j < ns; ++j) mx = fmaxf(mx, aL[wave][j]);
        float e[8], s = 0.0f;
        for (int j = 0; j < ns; ++j) { e[j] = __expf(aL[wave][j] - mx); s += e[j]; }
        float inv = 1.0f / s;
        for (int j = 0; j < ns; ++j) {
            float a = e[j] * inv + 1e-8f;
            attn[((size_t)bb * Q + tok) * ns + j] = a;
            aL[wave][j] = a;
        }
    }
    __syncthreads();
    if ((int)threadIdx.x < ns) {
        float s = 0.0f;
        for (int w = 0; w < 8; ++w) s += aL[w][threadIdx.x];
        atomicAdd(&colsum[bb * ns + threadIdx.x], s);
    }
}

// upd_raw[b*ns+j][d] += sum_{tok in chunk} attn[b,tok,j]*v[b,tok,d]
__global__ void attn_upd_kernel(const float* __restrict__ attn, const float* __restrict__ v,
                                float* __restrict__ upd_raw, int ns) {
    int row = blockIdx.x;          // bb*ns + j
    int bb = row / ns, j = row % ns;
    int chunk = blockIdx.y;        // 16 chunks of 576 tokens
    int d = threadIdx.x;
    const float* vb = v + (size_t)bb * Q * CH;
    const float* ab = attn + (size_t)bb * Q * ns + j;
    float acc = 0.0f;
    int t0 = chunk * 576, t1 = t0 + 576;
    for (int t = t0; t < t1; ++t)
        acc += ab[(size_t)t * ns] * vb[(size_t)t * CH + d];
    atomicAdd(&upd_raw[(size_t)row * CH + d], acc);
}

// GRU update + LN + MLP(256->512->256) residual, in-place on slots row n
__global__ void gru_mlp_kernel(const float* __restrict__ upd_raw, const float* __restrict__ colsum,
                               float* __restrict__ slots,
                               const float* __restrict__ wih, const float* __restrict__ whh,
                               const float* __restrict__ bih, const float* __restrict__ bhh,
                               const float* __restrict__ nm_g, const float* __restrict__ nm_b,
                               const float* __restrict__ w1, const float* __restrict__ b1,
                               const float* __restrict__ w2, const float* __restrict__ b2) {
    int n = blockIdx.x, d = threadIdx.x;
    __shared__ float X[256], Hp[256], Sm[256], Hid[512], red[256];
    float inv = 1.0f / colsum[n];
    X[d]  = upd_raw[(size_t)n * CH + d] * inv;
    Hp[d] = slots[(size_t)n * CH + d];
    __syncthreads();
    float gi0 = bih[d], gi1 = bih[256 + d], gi2 = bih[512 + d];
    float gh0 = bhh[d], gh1 = bhh[256 + d], gh2 = bhh[512 + d];
    const float* wi0 = wih + (size_t)d * CH;
    const float* wi1 = wih + (size_t)(256 + d) * CH;
    const float* wi2 = wih + (size_t)(512 + d) * CH;
    const float* wh0 = whh + (size_t)d * CH;
    const float* wh1 = whh + (size_t)(256 + d) * CH;
    const float* wh2 = whh + (size_t)(512 + d) * CH;
    for (int i = 0; i < CH; ++i) {
        float xv = X[i], hv = Hp[i];
        gi0 += xv * wi0[i]; gi1 += xv * wi1[i]; gi2 += xv * wi2[i];
        gh0 += hv * wh0[i]; gh1 += hv * wh1[i]; gh2 += hv * wh2[i];
    }
    float r  = 1.0f / (1.0f + __expf(-(gi0 + gh0)));
    float z  = 1.0f / (1.0f + __expf(-(gi1 + gh1)));
    float nn = tanhf(gi2 + r * gh2);
    float hnew = (1.0f - z) * nn + z * Hp[d];
    // LayerNorm(hnew) -> Sm
    red[d] = hnew; __syncthreads();
    for (int s = 128; s; s >>= 1) { if (d < s) red[d] += red[d + s]; __syncthreads(); }
    float mean = red[0] * (1.0f / 256.0f); __syncthreads();
    float dd = hnew - mean;
    red[d] = dd * dd; __syncthreads();
    for (int s = 128; s; s >>= 1) { if (d < s) red[d] += red[d + s]; __syncthreads(); }
    float var = red[0] * (1.0f / 256.0f);
    Sm[d] = dd * rsqrtf(var + 1e-5f) * nm_g[d] + nm_b[d];
    __syncthreads();
    float h0 = b1[d], h1 = b1[256 + d];
    const float* w1a = w1 + (size_t)d * CH;
    const float* w1b = w1 + (size_t)(256 + d) * CH;
    for (int i = 0; i < CH; ++i) { float s = Sm[i]; h0 += s * w1a[i]; h1 += s * w1b[i]; }
    Hid[d] = fmaxf(h0, 0.0f); Hid[256 + d] = fmaxf(h1, 0.0f);
    __syncthreads();
    float o = b2[d];
    const float* w2r = w2 + (size_t)d * 512;
    for (int i = 0; i < 512; ++i) o += Hid[i] * w2r[i];
    slots[(size_t)n * CH + d] = hnew + o;
}

// decode (collapsed): out_nhwc[bb][t][ch] = Wsum[o]*slots[4*i1+bb][t%256] + P[o][t]
__global__ void decode_kernel(const float* __restrict__ slots, const float* __restrict__ Wsum,
                              const float* __restrict__ P, float* __restrict__ out_nhwc,
                              int opn) {
    int t = blockIdx.x, bb = blockIdx.y, ch = threadIdx.x;
    int i1 = ch / opn, o = ch % opn;
    int n = 4 * i1 + bb;
    float val = Wsum[o] * slots[(size_t)n * CH + (t & 255)] + P[(size_t)o * Q + t];
    out_nhwc[((size_t)bb * Q + t) * CH + ch] = val;
}

__global__ void zero_kernel(float* __restrict__ p, int n) {
    int i = blockIdx.x * 256 + threadIdx.x;
    if (i < n) p[i] = 0.0f;
}

// ---------------------------------------------------------------------------
// Final gate: WMMA GEMM K=512 over concat(slots1,slots2) NHWC rows, fused
// sigmoid + blend epilogue writing NCHW output. Wave tile 32x64; the K loop
// is split into two branch-free 256-phases (slots1 then slots2).
// ---------------------------------------------------------------------------
__global__ void gate_final_kernel(const float* __restrict__ s1, const float* __restrict__ s2,
                                  const float* __restrict__ gw, const float* __restrict__ gb,
                                  const float* __restrict__ img, float* __restrict__ out) {
    int lane = threadIdx.x & 31, wave = threadIdx.x >> 5;
    int rstrip = wave & 1, ngrp = wave >> 1;
    int bb = blockIdx.y;
    int m0 = blockIdx.x * 64 + rstrip * 32;   // token base
    int n0 = ngrp * 64;                        // out-channel base
    int lm = lane & 15;
    int lk = (lane >> 4) * 2;
    const float* s1b = s1 + (size_t)bb * Q * CH;
    const float* s2b = s2 + (size_t)bb * Q * CH;
    v8f acc[2][4] = {};
    const float* A10 = s1b + (size_t)(m0 + lm) * CH;
    const float* A11 = s1b + (size_t)(m0 + 16 + lm) * CH;
    const float* A20 = s2b + (size_t)(m0 + lm) * CH;
    const float* A21 = s2b + (size_t)(m0 + 16 + lm) * CH;
    const float* W0 = gw + (size_t)(n0 + lm) * 512;
    // phase 1: k in [0,256) from slots1
    for (int kb = 0; kb < 256; kb += 4) {
        v2f a0 = *(const v2f*)(A10 + kb + lk);
        v2f a1 = *(const v2f*)(A11 + kb + lk);
#pragma unroll
        for (int nt = 0; nt < 4; ++nt) {
            v2f b = *(const v2f*)(W0 + (size_t)nt * 16 * 512 + kb + lk);
            acc[0][nt] = __builtin_amdgcn_wmma_f32_16x16x4_f32(
                false, a0, false, b, (short)0, acc[0][nt], false, false);
            acc[1][nt] = __builtin_amdgcn_wmma_f32_16x16x4_f32(
                false, a1, false, b, (short)0, acc[1][nt], false, false);
        }
    }
    // phase 2: k in [256,512) from slots2
    for (int kb = 0; kb < 256; kb += 4) {
        v2f a0 = *(const v2f*)(A20 + kb + lk);
        v2f a1 = *(const v2f*)(A21 + kb + lk);
#pragma unroll
        for (int nt = 0; nt < 4; ++nt) {
            v2f b = *(const v2f*)(W0 + (size_t)nt * 16 * 512 + 256 + kb + lk);
            acc[0][nt] = __builtin_amdgcn_wmma_f32_16x16x4_f32(
                false, a0, false, b, (short)0, acc[0][nt], false, false);
            acc[1][nt] = __builtin_amdgcn_wmma_f32_16x16x4_f32(
                false, a1, false, b, (short)0, acc[1][nt], false, false);
        }
    }
    int mq = (lane >> 4) * 8;
    const float* imgb = img + (size_t)bb * IMG_B;
    float* outb = out + (size_t)bb * IMG_B;
#pragma unroll
    for (int ms = 0; ms < 2; ++ms) {
        int mbase = m0 + ms * 16 + mq;
#pragma unroll
        for (int nt = 0; nt < 4; ++nt) {
            int o = n0 + nt * 16 + lm;
            float bv = gb[o];
#pragma unroll
            for (int v = 0; v < 8; ++v) {
                int t = mbase + v;
                float al = 1.0f / (1.0f + __expf(-(acc[ms][nt][v] + bv)));
                float a1 = s1b[(size_t)t * CH + o];
                float a2 = s2b[(size_t)t * CH + o];
                outb[(size_t)o * Q + t] = imgb[(size_t)o * Q + t] + al * a1 + (1.0f - al) * a2;
            }
        }
    }
}

// ---------------------------------------------------------------------------
// Host orchestration
// ---------------------------------------------------------------------------
enum {
    LN_G = 0, LN_Bi, NI_G, NI_Bi, NS_Gi, NS_Bi, NM_G, NM_Bi,
    MLP1_W1, MLP1_B1, MLP1_W2, MLP1_B2, MU, LOG_SIGMA, WK, WQ,
    MLP2_W1, MLP2_B1, MLP2_W2, MLP2_B2,
    GRU_WIH, GRU_WHH, GRU_BIH, GRU_BHH,
    ENC_W1, ENC_B1, ENC_W2, ENC_B2, DEC_W1, DEC_B1, DEC_W2, DEC_B2, WCONV,
    NPARAM
};

extern "C" void kernel_launch(void* const* d_in, const int* in_sizes, int n_in,
                              void* d_out, int out_size, void* d_ws, size_t ws_size,
                              hipStream_t stream) {
    (void)in_sizes; (void)n_in; (void)out_size; (void)ws_size;
    const float* image  = (const float*)d_in[0];
    const float* noise1 = (const float*)d_in[1];
    const float* noise2 = (const float*)d_in[2];
    const float* p1[NPARAM]; const float* p2[NPARAM];
    for (int i = 0; i < NPARAM; ++i) p1[i] = (const float*)d_in[3 + i];
    for (int i = 0; i < NPARAM; ++i) p2[i] = (const float*)d_in[3 + NPARAM + i];
    const float* gate_w = (const float*)d_in[3 + 2 * NPARAM];
    const float* gate_b = (const float*)d_in[4 + 2 * NPARAM];
    float* out = (float*)d_out;

    // workspace bump allocator (256B aligned)
    char* wp = (char*)d_ws;
    auto alloc = [&](size_t nfloats) {
        float* r = (float*)wp;
        wp += ((nfloats * sizeof(float) + 255) / 256) * 256;
        return r;
    };
    float* enc_pos = alloc((size_t)Q * CH);
    float* dec_pos = alloc((size_t)Q * CH);
    float* Pbuf    = alloc((size_t)64 * Q);    // max opn = 64
    float* Wsum    = alloc(64);
    float* bufA    = alloc((size_t)BT * CH);
    float* bufB    = alloc((size_t)BT * CH);   // also serves as kbuf (k == v)
    float* bufC    = alloc((size_t)BT * CH);
    float* s1buf   = alloc((size_t)BT * CH);   // slots1 decoded, NHWC
    float* s2buf   = alloc((size_t)BT * CH);   // slots2 decoded, NHWC
    float* slots   = alloc((size_t)NB * 8 * CH);
    float* qb      = alloc((size_t)NB * 8 * CH);
    float* attnb   = alloc((size_t)NB * Q * 8);
    float* colsum  = alloc(64);
    float* updb    = alloc((size_t)NB * 8 * CH);

    auto run_module = [&](const float* const* p, int ns, const float* noise,
                          const float* gather_add, float* out_nhwc) {
        int opn = CH / ns;
        // positional embeddings + decoder precompute
        pos_mlp_kernel<<<Q, 256, 0, stream>>>(p[ENC_W1], p[ENC_B1], p[ENC_W2], p[ENC_B2], enc_pos);
        pos_mlp_kernel<<<Q, 256, 0, stream>>>(p[DEC_W1], p[DEC_B1], p[DEC_W2], p[DEC_B2], dec_pos);
        wsum_kernel<<<1, opn, 0, stream>>>(p[WCONV], Wsum);
        pcomp_kernel<<<dim3(Q / 256, opn), 256, 0, stream>>>(p[WCONV], dec_pos, Pbuf);
        // encoder: (x [+ slots1] + pos) -> LN -> MLP -> LN -> k
        if (gather_add)
            ln_kernel<true, true><<<BT, 256, 0, stream>>>(image, gather_add, enc_pos,
                                                          p[LN_G], p[LN_Bi], bufA);
        else
            ln_kernel<false, true><<<BT, 256, 0, stream>>>(image, nullptr, enc_pos,
                                                           p[LN_G], p[LN_Bi], bufA);
        gemm_kernel<1, 1><<<BT / 64, 256, 0, stream>>>(bufA, p[MLP1_W1], p[MLP1_B1], bufB);
        gemm_kernel<0, 1><<<BT / 64, 256, 0, stream>>>(bufB, p[MLP1_W2], p[MLP1_B2], bufC);
        ln_kernel<false, false><<<BT, 256, 0, stream>>>(bufC, nullptr, nullptr,
                                                        p[NI_G], p[NI_Bi], bufA);
        gemm_kernel<0, 0><<<BT / 64, 256, 0, stream>>>(bufA, p[WK], nullptr, bufB); // k (== v)
        // slot attention iterations
        slots_init_kernel<<<NB * ns, 256, 0, stream>>>(p[MU], p[LOG_SIGMA], noise, slots, ns);
        for (int it = 0; it < 3; ++it) {
            zero_kernel<<<1, 64, 0, stream>>>(colsum, 64);
            zero_kernel<<<NB * ns, 256, 0, stream>>>(updb, NB * ns * CH);
            slot_q_kernel<<<NB * ns, 256, 0, stream>>>(slots, p[NS_Gi], p[NS_Bi], p[WQ], qb);
            attn_logits_kernel<<<dim3(Q / 8, NB), 256, 0, stream>>>(bufB, qb, attnb, colsum, ns);
            attn_upd_kernel<<<dim3(NB * ns, 16), 256, 0, stream>>>(attnb, bufB, updb, ns);
            gru_mlp_kernel<<<NB * ns, 256, 0, stream>>>(updb, colsum, slots,
                p[GRU_WIH], p[GRU_WHH], p[GRU_BIH], p[GRU_BHH],
                p[NM_G], p[NM_Bi], p[MLP2_W1], p[MLP2_B1], p[MLP2_W2], p[MLP2_B2]);
        }
        decode_kernel<<<dim3(Q, NB), 256, 0, stream>>>(slots, Wsum, Pbuf, out_nhwc, opn);
    };

    run_module(p1, 8, noise1, nullptr, s1buf);
    run_module(p2, 4, noise2, s1buf, s2buf);
    gate_final_kernel<<<dim3(Q / 64, NB), 256, 0, stream>>>(s1buf, s2buf, gate_w, gate_b,
                                                            image, out);
}